// WasserteinFlow_89000312307793
// MI455X (gfx1250) — compile-verified
//
#include <hip/hip_runtime.h>
#include <math.h>

#define G_GROUPS 128
#define N_IN     (G_GROUPS * 4)

#define SQRT2_F        1.41421356237309515f
#define INV_SQRT_2PI_F 0.3989422804014327f

// ---- workspace layout (floats) ----
#define WS_TV   0                        // [G,4] time MLP output (z-independent)
#define WS_CT   (WS_TV + G_GROUPS * 4)   // [G,4] ct = Wf @ d(tv)/dt
#define WS_DSDT (WS_CT + G_GROUPS * 4)   // [G]   sum_b dS/dt
#define WS_PART (WS_DSDT + G_GROUPS)     // [G,8] partial sums
#define WS_SCAL (WS_PART + G_GROUPS * 8) // 6 scalars from mu/sigma

__device__ __forceinline__ float t_tanh(float x) {
#if __has_builtin(__builtin_amdgcn_tanhf)
  return __builtin_amdgcn_tanhf(x);   // v_tanh_f32 (CDNA5 TRANS op, co-executes with VALU)
#else
  return tanhf(x);
#endif
}

// xor-lane-swap within wave32 via ds_swizzle group-of-32 mode:
// imm = {xor_mask[14:10], or_mask[9:5]=0, and_mask[4:0]=0x1f} -> i = lane ^ m
template <int M>
__device__ __forceinline__ float xr_add(float x) {
#if __has_builtin(__builtin_amdgcn_ds_swizzle)
  return x + __int_as_float(
      __builtin_amdgcn_ds_swizzle(__float_as_int(x), (M << 10) | 0x1f));
#else
  return x + __shfl_xor(x, M, 32);
#endif
}

// =====================================================================
// Kernel 1: per-group, z-independent precompute (time embedding MLP),
// plus the truncated-normal scalar constants.
// =====================================================================
__global__ __launch_bounds__(G_GROUPS) void k_group_pre(
    const float* __restrict__ time_step, const float* __restrict__ mu_p,
    const float* __restrict__ sigma_p,
    const float* __restrict__ Wt,  const float* __restrict__ bt,
    const float* __restrict__ Wt2, const float* __restrict__ bt2,
    const float* __restrict__ Wf,  float* __restrict__ ws)
{
  const int g = threadIdx.x;
  if (g == 0) {
    const float mu = mu_p[0], sg = sigma_p[0];
    const float sig2 = sg * sg + 1e-5f;
    const float pb = 0.5f * (1.0f + erff((10.0f  - mu) / sg / SQRT2_F));
    const float pa = 0.5f * (1.0f + erff((-10.0f - mu) / sg / SQRT2_F));
    ws[WS_SCAL + 0] = mu;
    ws[WS_SCAL + 1] = 1.0f / sg;                          // inv_sigma
    ws[WS_SCAL + 2] = 1.0f / sig2;                        // inv_sig2
    ws[WS_SCAL + 3] = sg * sg;                            // sigma^2 (plain)
    ws[WS_SCAL + 4] = 1.0f / (sg * sg * sg * sg + 1e-5f); // inv_sig4
    ws[WS_SCAL + 5] = 1.0f / (pb - pa);                   // inv_den
  }
  if (g >= G_GROUPS) return;

  const float t  = time_step[0];
  // half = 2: freqs = [1, exp(-log(1e4)/2)]
  const float f0 = 1.0f;
  const float f1 = __expf(-0.5f * __logf(10000.0f));
  const float te0 = t * f0, te1 = t * f1;
  const float temb[4]  = { sinf(te0), sinf(te1),  cosf(te0),  cosf(te1) };
  const float dtemb[4] = { f0 * cosf(te0),  f1 * cosf(te1),
                          -f0 * sinf(te0), -f1 * sinf(te1) };

  float at[4], dat[4];
#pragma unroll
  for (int o = 0; o < 4; ++o) {
    float a = bt[g * 4 + o], d = 0.f;
#pragma unroll
    for (int i = 0; i < 4; ++i) {
      const float w = Wt[g * 16 + o * 4 + i];
      a += w * temb[i];
      d += w * dtemb[i];
    }
    at[o] = a; dat[o] = d;
  }
  float ge[4], dge[4];
#pragma unroll
  for (int o = 0; o < 4; ++o) {
    const float x   = at[o];
    const float Phi = 0.5f * (1.0f + erff(x * 0.7071067811865475f));
    const float phi = __expf(-0.5f * x * x) * INV_SQRT_2PI_F;
    ge[o]  = x * Phi;                  // exact GELU
    dge[o] = (Phi + x * phi) * dat[o]; // gelu'(a) * da/dt
  }
  float tv[4], dtv[4];
#pragma unroll
  for (int o = 0; o < 4; ++o) {
    float a = bt2[g * 4 + o], d = 0.f;
#pragma unroll
    for (int i = 0; i < 4; ++i) {
      const float w = Wt2[g * 16 + o * 4 + i];
      a += w * ge[i];
      d += w * dge[i];
    }
    tv[o] = a; dtv[o] = d;
  }
#pragma unroll
  for (int n = 0; n < 4; ++n) {
    float c = 0.f;
#pragma unroll
    for (int m = 0; m < 4; ++m) c += Wf[g * 16 + n * 4 + m] * dtv[m];
    ws[WS_TV + g * 4 + n] = tv[n];
    ws[WS_CT + g * 4 + n] = c;   // ct = Wf @ dtv  (so dS/dt = p * sum_n v_n ct_n)
  }
}

// =====================================================================
// Kernel 2: dS/dt partial batch-sums per group (deterministic).
// Wave-level reduction via V_WMMA_F32_16X16X4_F32:
//   A = all-ones 16x4 (layout-invariant), B = {partial, 0} per lane
//   -> every row of D = column sums of B.  Row M=0 sits in VGPR0 lanes 0-15,
//   row M=8 (identical values) in lanes 16-31, so an xor-reduce over masks
//   {1,2,4,8} (stays inside each 16-lane half) leaves the full wave sum in
//   every lane.  4 ds_swizzle steps, no scaling.
// =====================================================================
__global__ __launch_bounds__(256) void k_dsdt(
    const float* __restrict__ z,
    const float* __restrict__ Wx, const float* __restrict__ bx,
    const float* __restrict__ Wf, const float* __restrict__ bf,
    const float* __restrict__ Wo, const float* __restrict__ bo,
    float* __restrict__ ws, int B)
{
  const int g    = blockIdx.x;   // group (uniform -> weights live in SGPRs)
  const int part = blockIdx.y;   // 8 batch partitions

  float wx[16], wf[16], wo[4], vbx[4], vbf[4], tv[4], ct[4];
#pragma unroll
  for (int i = 0; i < 16; ++i) { wx[i] = Wx[g * 16 + i]; wf[i] = Wf[g * 16 + i]; }
#pragma unroll
  for (int i = 0; i < 4; ++i) {
    wo[i]  = Wo[g * 4 + i];  vbx[i] = bx[g * 4 + i];  vbf[i] = bf[g * 4 + i];
    tv[i]  = ws[WS_TV + g * 4 + i];  ct[i] = ws[WS_CT + g * 4 + i];
  }
  const float sbo = bo[g];

  const int Bq    = B >> 3;            // batch elems per partition
  const int iters = Bq >> 8;           // 256 threads
  const int b0    = part * Bq + (int)threadIdx.x;
  float acc = 0.f;
  for (int it = 0; it < iters; ++it) { // uniform trip count: EXEC stays all-ones
    const int b = b0 + (it << 8);
    const float4 z4 = *(const float4*)(z + (size_t)b * N_IN + g * 4);
    if (it + 1 < iters)                // scalar condition -> s_cbranch, in-bounds only
      __builtin_prefetch(z + (size_t)(b + 256) * N_IN + g * 4, 0, 2); // global_prefetch_b8
    const float zz[4] = { z4.x, z4.y, z4.z, z4.w };
    float x[4], h[4];
#pragma unroll
    for (int o = 0; o < 4; ++o) {
      float a = vbx[o];
#pragma unroll
      for (int i = 0; i < 4; ++i) a += wx[o * 4 + i] * zz[i];
      x[o] = t_tanh(a);
    }
#pragma unroll
    for (int n = 0; n < 4; ++n) {
      float a = vbf[n];
#pragma unroll
      for (int m = 0; m < 4; ++m) a += wf[n * 4 + m] * (x[m] + tv[m]);
      h[n] = t_tanh(a);
    }
    float a3 = sbo;
#pragma unroll
    for (int n = 0; n < 4; ++n) a3 += wo[n] * h[n];
    const float S = t_tanh(a3);
    const float p = 1.0f - S * S;
    float d = 0.f;
#pragma unroll
    for (int n = 0; n < 4; ++n) d += wo[n] * (1.0f - h[n] * h[n]) * ct[n];
    acc += p * d;
  }

  // ---- wave reduction (EXEC all-ones here: uniform control flow) ----
  float wsum;
#if __has_builtin(__builtin_amdgcn_wmma_f32_16x16x4_f32)
  {
    typedef __attribute__((ext_vector_type(2))) float v2f;
    typedef __attribute__((ext_vector_type(8))) float v8f;
    v2f Aone = { 1.0f, 1.0f };
    v2f Bval = { acc,  0.0f };
    v8f Cacc = { 0.f, 0.f, 0.f, 0.f, 0.f, 0.f, 0.f, 0.f };
    Cacc = __builtin_amdgcn_wmma_f32_16x16x4_f32(
        false, Aone, false, Bval, (short)0, Cacc, false, false);
    wsum = Cacc[0];            // 16 col-sums, mirrored across the two lane halves
    wsum = xr_add<1>(wsum);
    wsum = xr_add<2>(wsum);
    wsum = xr_add<4>(wsum);
    wsum = xr_add<8>(wsum);    // full wave sum in every lane
  }
#else
  wsum = acc;
  wsum = xr_add<1>(wsum);
  wsum = xr_add<2>(wsum);
  wsum = xr_add<4>(wsum);
  wsum = xr_add<8>(wsum);
  wsum = xr_add<16>(wsum);
#endif

  __shared__ float red[8];
  const int wave = threadIdx.x >> 5;
  if ((threadIdx.x & 31) == 0) red[wave] = wsum;
  __syncthreads();
  if (threadIdx.x == 0) {
    float s = 0.f;
#pragma unroll
    for (int i = 0; i < 8; ++i) s += red[i];   // fixed order -> deterministic
    ws[WS_PART + g * 8 + part] = s;
  }
}

__global__ __launch_bounds__(G_GROUPS) void k_dsdt_reduce(float* __restrict__ ws)
{
  const int g = threadIdx.x;
  if (g < G_GROUPS) {
    float s = 0.f;
#pragma unroll
    for (int p = 0; p < 8; ++p) s += ws[WS_PART + g * 8 + p];
    ws[WS_DSDT + g] = s;
  }
}

// =====================================================================
// Kernel 3: main — analytic S, grad, Hessian, and the four outputs.
// H = -2p [ S q q^T + Wx^T diag(c) Wx + F^T diag(s) F ],  F = Wf Dx Wx
// =====================================================================
__global__ __launch_bounds__(256) void k_main(
    const float* __restrict__ z,
    const float* __restrict__ Wx, const float* __restrict__ bx,
    const float* __restrict__ Wf, const float* __restrict__ bf,
    const float* __restrict__ Wo, const float* __restrict__ bo,
    const float* __restrict__ beta, const float* __restrict__ ws,
    float* __restrict__ out, int B)
{
  const int g = blockIdx.y;                       // uniform per block
  const int b = blockIdx.x * 256 + threadIdx.x;

  // ---- uniform per-group constants (SGPR-resident) ----
  float wx[16], wf[16], wo[4], vbx[4], vbf[4], tv[4];
#pragma unroll
  for (int i = 0; i < 16; ++i) { wx[i] = Wx[g * 16 + i]; wf[i] = Wf[g * 16 + i]; }
#pragma unroll
  for (int i = 0; i < 4; ++i) {
    wo[i]  = Wo[g * 4 + i];  vbx[i] = bx[g * 4 + i];  vbf[i] = bf[g * 4 + i];
    tv[i]  = ws[WS_TV + g * 4 + i];
  }
  const float sbo  = bo[g];
  const float bb   = beta[g];
  const float dsum = ws[WS_DSDT + g];
  const float mu        = ws[WS_SCAL + 0];
  const float inv_sigma = ws[WS_SCAL + 1];
  const float inv_sig2  = ws[WS_SCAL + 2];
  const float sigma2    = ws[WS_SCAL + 3];
  const float inv_sig4  = ws[WS_SCAL + 4];
  const float inv_den   = ws[WS_SCAL + 5];

  const float4 z4 = *(const float4*)(z + (size_t)b * N_IN + g * 4);
  const float zz[4] = { z4.x, z4.y, z4.z, z4.w };

  // ---- truncated-normal density, nabla_log_phi, delta_phi_over_phi ----
  float rho[4], nlp[4], dpp[4];
  float lsum = 0.f;
#pragma unroll
  for (int k = 0; k < 4; ++k) {
    const float zk = zz[k];
    float xs = (zk > 10.0f) ? zk + 100.0f : zk;
    xs       = (xs < -10.0f) ? xs - 100.0f : xs;
    const float xn = (xs - mu) * inv_sigma;
    float dens = __expf(-0.5f * xn * xn) * INV_SQRT_2PI_F;
    dens = (xn > 10.0f || xn < -10.0f) ? 0.0f : dens;
    rho[k] = dens * inv_den + 1e-10f;
    lsum  += __logf(rho[k]);
    const float zm = zk - mu;
    nlp[k] = -zm * inv_sig2;
    dpp[k] = (zm * zm - sigma2) * inv_sig4;
  }

  // ---- forward ----
  float x[4], h[4];
#pragma unroll
  for (int o = 0; o < 4; ++o) {
    float a = vbx[o];
#pragma unroll
    for (int i = 0; i < 4; ++i) a += wx[o * 4 + i] * zz[i];
    x[o] = t_tanh(a);
  }
#pragma unroll
  for (int n = 0; n < 4; ++n) {
    float a = vbf[n];
#pragma unroll
    for (int m = 0; m < 4; ++m) a += wf[n * 4 + m] * (x[m] + tv[m]);
    h[n] = t_tanh(a);
  }
  float a3 = sbo;
#pragma unroll
  for (int n = 0; n < 4; ++n) a3 += wo[n] * h[n];
  const float S = t_tanh(a3);
  const float p = 1.0f - S * S;

  // ---- first derivative chain ----
  float dh[4], v[4], dx[4], u1[4], w[4], q[4], sn[4], c[4];
#pragma unroll
  for (int n = 0; n < 4; ++n) { dh[n] = 1.0f - h[n] * h[n]; v[n] = wo[n] * dh[n]; }
#pragma unroll
  for (int m = 0; m < 4; ++m) {
    float a = 0.f;
#pragma unroll
    for (int n = 0; n < 4; ++n) a += wf[n * 4 + m] * v[n];
    u1[m] = a;
    dx[m] = 1.0f - x[m] * x[m];
    w[m]  = dx[m] * u1[m];
  }
#pragma unroll
  for (int j = 0; j < 4; ++j) {
    float a = 0.f;
#pragma unroll
    for (int m = 0; m < 4; ++m) a += wx[m * 4 + j] * w[m];
    q[j] = a;
  }
#pragma unroll
  for (int n = 0; n < 4; ++n) sn[n] = h[n] * v[n];          // h * Wo * (1-h^2)
#pragma unroll
  for (int m = 0; m < 4; ++m) c[m] = x[m] * dx[m] * u1[m];

  // ---- F = Wf * Dx * Wx ----
  float F[16];
#pragma unroll
  for (int n = 0; n < 4; ++n)
#pragma unroll
    for (int k = 0; k < 4; ++k) {
      float a = 0.f;
#pragma unroll
      for (int m = 0; m < 4; ++m) a += wf[n * 4 + m] * dx[m] * wx[m * 4 + k];
      F[n * 4 + k] = a;
    }

  // ---- Hessian (symmetric) ----
  float H[16];
#pragma unroll
  for (int j = 0; j < 4; ++j)
#pragma unroll
    for (int k = j; k < 4; ++k) {
      float t3 = 0.f, t2 = 0.f;
#pragma unroll
      for (int n = 0; n < 4; ++n) t3 += sn[n] * F[n * 4 + j] * F[n * 4 + k];
#pragma unroll
      for (int m = 0; m < 4; ++m) t2 += c[m] * wx[m * 4 + j] * wx[m * 4 + k];
      const float val = -2.0f * p * (S * q[j] * q[k] + t2 + t3);
      H[j * 4 + k] = val;
      H[k * 4 + j] = val;
    }

  float colsum[4], corr[4];
#pragma unroll
  for (int k = 0; k < 4; ++k) {
    float cs = 0.f, cr = 0.f;
#pragma unroll
    for (int j = 0; j < 4; ++j) { cs += H[j * 4 + k]; cr += rho[j] * H[j * 4 + k]; }
    colsum[k] = cs; corr[k] = cr;
  }

  // ---- outputs ----
  const float u = S - bb * lsum;                   // S - beta*log(prod rho)
  float pde[4], uz[4], uzz[4];
#pragma unroll
  for (int k = 0; k < 4; ++k) {
    uz[k]  = p * q[k] - bb * nlp[k];
    const float ut = dsum + bb * corr[k] / (rho[k] + 1e-5f);
    uzz[k] = colsum[k] + bb * inv_sig2;            // H colsum - beta*(-1/sig2)
    pde[k] = ut + 0.5f * uz[k] * uz[k]
           + 0.125f * bb * bb * (nlp[k] * nlp[k] - 2.0f * dpp[k]);
  }

  const size_t base = (size_t)b * N_IN + g * 4;
  const size_t SZ   = (size_t)B * N_IN;
  *(float4*)(out + base)                       = make_float4(pde[0], pde[1], pde[2], pde[3]);
  *(float4*)(out + SZ + base)                  = make_float4(uz[0],  uz[1],  uz[2],  uz[3]);
  out[2 * SZ + (size_t)b * G_GROUPS + g]       = u;
  *(float4*)(out + 2 * SZ + (size_t)B * G_GROUPS + base)
                                               = make_float4(uzz[0], uzz[1], uzz[2], uzz[3]);
}

// =====================================================================
extern "C" void kernel_launch(void* const* d_in, const int* in_sizes, int n_in,
                              void* d_out, int out_size, void* d_ws, size_t ws_size,
                              hipStream_t stream)
{
  (void)n_in; (void)out_size; (void)ws_size;
  const float* z    = (const float*)d_in[0];
  const float* ts   = (const float*)d_in[1];
  const float* mu   = (const float*)d_in[2];
  const float* sig  = (const float*)d_in[3];
  const float* Wx   = (const float*)d_in[4];
  const float* bx   = (const float*)d_in[5];
  const float* Wt   = (const float*)d_in[6];
  const float* bt   = (const float*)d_in[7];
  const float* Wt2  = (const float*)d_in[8];
  const float* bt2  = (const float*)d_in[9];
  const float* Wf   = (const float*)d_in[10];
  const float* bf   = (const float*)d_in[11];
  const float* Wo   = (const float*)d_in[12];
  const float* bo   = (const float*)d_in[13];
  const float* beta = (const float*)d_in[14];
  float* out = (float*)d_out;
  float* ws  = (float*)d_ws;

  const int B = in_sizes[0] / N_IN;   // 8192

  k_group_pre<<<1, G_GROUPS, 0, stream>>>(ts, mu, sig, Wt, bt, Wt2, bt2, Wf, ws);

  dim3 gds(G_GROUPS, 8);
  k_dsdt<<<gds, 256, 0, stream>>>(z, Wx, bx, Wf, bf, Wo, bo, ws, B);

  k_dsdt_reduce<<<1, G_GROUPS, 0, stream>>>(ws);

  dim3 gm(B / 256, G_GROUPS);
  k_main<<<gm, 256, 0, stream>>>(z, Wx, bx, Wf, bf, Wo, bo, beta, ws, out, B);
}